// Decoder_39109972197587
// MI455X (gfx1250) — compile-verified
//
#include <hip/hip_runtime.h>
#include <math.h>
#include <stddef.h>

// Problem constants (match reference)
#define T_LEN   50
#define BATCH   64
#define SRC_LEN 50
#define EMB     512
#define HID     512

#define NBLK    32      // persistent workgroups (32 n-tiles of 16 units)
#define NTHR    128     // 4 waves: m-tiles of 16 batch rows

typedef __attribute__((ext_vector_type(2))) float v2f;
typedef __attribute__((ext_vector_type(8))) float v8f;

// ---------------------------------------------------------------------------
// CDNA5 f32 WMMA: D(16x16) = A(16x4) * B(4x16) + C, full f32 precision.
// ---------------------------------------------------------------------------
__device__ __forceinline__ v8f wmma4(v2f a, v2f b, v8f c) {
  return __builtin_amdgcn_wmma_f32_16x16x4_f32(false, a, false, b,
                                               (short)0, c, false, false);
}

__device__ __forceinline__ v2f ld2(const float* p) {
  return *(const v2f*)p;   // 8B-aligned: all K offsets are even
}

__device__ __forceinline__ float sigf(float x) {
  return 1.0f / (1.0f + __expf(-x));
}
__device__ __forceinline__ float tanh_fast(float x) {
  float ax = fabsf(x);
  float e  = __expf(2.0f * ax);          // +inf for large ax -> t = 1
  float t  = 1.0f - 2.0f / (e + 1.0f);
  return copysignf(t, x);
}

// ---------------------------------------------------------------------------
// Device-wide generation barrier across NBLK resident workgroups.
// bar[0] = arrival count, bar[1] = generation. Agent-scope acq/rel atomics
// provide the cross-WGP cache maintenance (release producer stores, acquire
// invalidates stale near-cache lines before the next stage reads).
// ---------------------------------------------------------------------------
__device__ __forceinline__ void grid_barrier(unsigned* bar) {
  __syncthreads();
  if (threadIdx.x == 0) {
    unsigned* cnt = bar;
    unsigned* gen = bar + 1;
    unsigned g = __hip_atomic_load(gen, __ATOMIC_RELAXED, __HIP_MEMORY_SCOPE_AGENT);
    unsigned a = __hip_atomic_fetch_add(cnt, 1u, __ATOMIC_ACQ_REL, __HIP_MEMORY_SCOPE_AGENT);
    if (a == NBLK - 1) {
      // last to arrive: reset count, then release everyone
      __hip_atomic_store(cnt, 0u, __ATOMIC_RELAXED, __HIP_MEMORY_SCOPE_AGENT);
      __hip_atomic_fetch_add(gen, 1u, __ATOMIC_ACQ_REL, __HIP_MEMORY_SCOPE_AGENT);
    } else {
      while (__hip_atomic_load(gen, __ATOMIC_ACQUIRE, __HIP_MEMORY_SCOPE_AGENT) == g)
        __builtin_amdgcn_s_sleep(1);
    }
  }
  __syncthreads();
}

// ---------------------------------------------------------------------------
// Persistent decoder: whole T=50 scan in one kernel. Per step:
//   lstm0 -> bar -> lstm1 -> bar -> attn_in -> bar -> attn_ctx -> bar
//         -> attn_out -> bar
// Stage mapping for the WMMA GEMMs: blockIdx.x = 16-wide unit tile,
// wave (tid>>5) = 16-row batch tile; LSTM waves keep 4 gate accumulators
// (i,f,g,o) sharing one A fragment.
// ---------------------------------------------------------------------------
__global__ __launch_bounds__(NTHR, 1) void decoder_kernel(
    const int*   __restrict__ tokens,    // [T, BATCH]
    const float* __restrict__ emb,       // [V, EMB]
    const float* __restrict__ init_out,  // [BATCH, HID]
    const float* __restrict__ context,   // [SRC, BATCH, HID]
    const float* __restrict__ Wih0,      // [4H, EMB+HID]
    const float* __restrict__ Whh0,      // [4H, HID]
    const float* __restrict__ b0,        // [4H]
    const float* __restrict__ Wih1,      // [4H, HID]
    const float* __restrict__ Whh1,      // [4H, HID]
    const float* __restrict__ b1,        // [4H]
    const float* __restrict__ Wai,       // [HID, HID]
    const float* __restrict__ Wao,       // [HID, 2*HID]
    float* __restrict__ outputs,         // [T, BATCH, HID] (d_out)
    float* __restrict__ attnF,           // [BATCH, SRC]    (d_out, last step wins)
    float* __restrict__ h0s,             // [2, BATCH, HID] ping-pong
    float* __restrict__ c0s,
    float* __restrict__ h1s,
    float* __restrict__ c1s,
    float* __restrict__ tgt,             // [BATCH, HID]
    float* __restrict__ wctx,            // [BATCH, HID]
    unsigned* bar)                       // [2] count/gen
{
  const int tid  = threadIdx.x;
  const int lane = tid & 31;
  const int wv   = tid >> 5;             // m-tile
  const int hi   = lane >> 4;
  const int l16  = lane & 15;
  const int blk  = blockIdx.x;           // n-tile
  const int n0   = blk * 16;
  const int m    = wv * 16 + l16;        // batch row for A fragments
  const int ko   = 2 * hi;               // K sub-offset per ISA A/B layout
  const int col  = n0 + l16;             // unit column for B fragments
  const int S    = BATCH * HID;

  __shared__ float s_tgt[2][HID];
  __shared__ float s_sc[2][64];

  // Time-invariant weight fragment pointers / biases (hoisted out of scan)
  const float* bih0[4]; const float* bhh0[4];
  const float* bih1[4]; const float* bhh1[4];
  float bias0[4], bias1[4];
#pragma unroll
  for (int g = 0; g < 4; ++g) {
    const int row = g * HID + col;
    bih0[g] = Wih0 + (size_t)row * (EMB + HID) + ko;
    bhh0[g] = Whh0 + (size_t)row * HID + ko;
    bih1[g] = Wih1 + (size_t)row * HID + ko;
    bhh1[g] = Whh1 + (size_t)row * HID + ko;
    bias0[g] = b0[row];
    bias1[g] = b1[row];
  }
  const float* bai = Wai + (size_t)col * HID + ko;
  const float* bao = Wao + (size_t)col * (2 * HID) + ko;

  for (int t = 0; t < T_LEN; ++t) {
    const int rd = t & 1, wr = rd ^ 1;
    const float* prev = (t == 0) ? init_out : outputs + (size_t)(t - 1) * S;
    const float* h0p = h0s + rd * S;  float* h0n = h0s + wr * S;
    const float* c0p = c0s + rd * S;  float* c0n = c0s + wr * S;
    const float* h1p = h1s + rd * S;  float* h1n = h1s + wr * S;
    const float* c1p = c1s + rd * S;  float* c1n = c1s + wr * S;

    // ================= stage 1: LSTM layer 0 =================
    {
      const int tk = tokens[t * BATCH + m];
      const float* a0 = emb  + (size_t)tk * EMB + ko;   // embedding gather
      const float* a1 = prev + m * HID + ko;            // input feed
      const float* a2 = h0p  + m * HID + ko;            // recurrent h

      v8f acc[4] = {};
#pragma unroll 4
      for (int k = 0; k < EMB; k += 4) {
        v2f a = ld2(a0 + k);
#pragma unroll
        for (int g = 0; g < 4; ++g) acc[g] = wmma4(a, ld2(bih0[g] + k), acc[g]);
      }
#pragma unroll 4
      for (int k = 0; k < HID; k += 4) {
        v2f a = ld2(a1 + k);
#pragma unroll
        for (int g = 0; g < 4; ++g) acc[g] = wmma4(a, ld2(bih0[g] + EMB + k), acc[g]);
      }
#pragma unroll 4
      for (int k = 0; k < HID; k += 4) {
        v2f a = ld2(a2 + k);
#pragma unroll
        for (int g = 0; g < 4; ++g) acc[g] = wmma4(a, ld2(bhh0[g] + k), acc[g]);
      }
      // C/D layout: element (vgpr r, lane) -> M = r + 8*hi, N = l16
#pragma unroll
      for (int r = 0; r < 8; ++r) {
        const int b = wv * 16 + r + 8 * hi;
        const float iv = sigf(acc[0][r] + bias0[0]);
        const float fv = sigf(acc[1][r] + bias0[1]);
        const float gv = tanh_fast(acc[2][r] + bias0[2]);
        const float ov = sigf(acc[3][r] + bias0[3]);
        const float cp = c0p[b * HID + col];
        const float cn = fv * cp + iv * gv;
        c0n[b * HID + col] = cn;
        h0n[b * HID + col] = ov * tanh_fast(cn);
      }
    }
    grid_barrier(bar);

    // ================= stage 2: LSTM layer 1 =================
    {
      const float* a0 = h0n + m * HID + ko;
      const float* a1 = h1p + m * HID + ko;

      v8f acc[4] = {};
#pragma unroll 4
      for (int k = 0; k < HID; k += 4) {
        v2f a = ld2(a0 + k);
#pragma unroll
        for (int g = 0; g < 4; ++g) acc[g] = wmma4(a, ld2(bih1[g] + k), acc[g]);
      }
#pragma unroll 4
      for (int k = 0; k < HID; k += 4) {
        v2f a = ld2(a1 + k);
#pragma unroll
        for (int g = 0; g < 4; ++g) acc[g] = wmma4(a, ld2(bhh1[g] + k), acc[g]);
      }
#pragma unroll
      for (int r = 0; r < 8; ++r) {
        const int b = wv * 16 + r + 8 * hi;
        const float iv = sigf(acc[0][r] + bias1[0]);
        const float fv = sigf(acc[1][r] + bias1[1]);
        const float gv = tanh_fast(acc[2][r] + bias1[2]);
        const float ov = sigf(acc[3][r] + bias1[3]);
        const float cp = c1p[b * HID + col];
        const float cn = fv * cp + iv * gv;
        c1n[b * HID + col] = cn;
        h1n[b * HID + col] = ov * tanh_fast(cn);
      }
    }
    grid_barrier(bar);

    // ================= stage 3: tgt = rnn_out @ Wai^T =================
    {
      const float* a = h1n + m * HID + ko;
      v8f acc = {};
#pragma unroll 8
      for (int k = 0; k < HID; k += 4)
        acc = wmma4(ld2(a + k), ld2(bai + k), acc);
#pragma unroll
      for (int r = 0; r < 8; ++r)
        tgt[(wv * 16 + r + 8 * hi) * HID + col] = acc[r];
    }
    grid_barrier(bar);

    // ============ stage 4: scores -> softmax -> weighted context ============
    // Each block handles 2 batch rows (NBLK*2 == BATCH); 2 waves per batch.
    {
      const int base = blk * 2 * HID;
      for (int i = tid; i < 2 * HID; i += NTHR)
        s_tgt[i >> 9][i & (HID - 1)] = tgt[base + i];
      __syncthreads();

      const int bb  = wv >> 1;        // which of the two batches
      const int sub = wv & 1;         // sub-wave within the batch pair
      const int b   = blk * 2 + bb;

      for (int s = sub; s < SRC_LEN; s += 2) {
        const float* cp = context + ((size_t)s * BATCH + b) * HID;
        float p = 0.f;
        for (int h = lane; h < HID; h += 32) p += cp[h] * s_tgt[bb][h];
#pragma unroll
        for (int off = 16; off; off >>= 1) p += __shfl_down(p, off, 32);
        if (lane == 0) s_sc[bb][s] = p;
      }
      __syncthreads();

      if (sub == 0) {                 // waves 0 & 2: softmax over 50 scores
        float v0 = (lane < SRC_LEN)      ? s_sc[bb][lane]      : -INFINITY;
        float v1 = (lane + 32 < SRC_LEN) ? s_sc[bb][lane + 32] : -INFINITY;
        float mx = fmaxf(v0, v1);
#pragma unroll
        for (int off = 16; off; off >>= 1) mx = fmaxf(mx, __shfl_xor(mx, off, 32));
        float e0 = (lane < SRC_LEN)      ? __expf(v0 - mx) : 0.f;
        float e1 = (lane + 32 < SRC_LEN) ? __expf(v1 - mx) : 0.f;
        float sm = e0 + e1;
#pragma unroll
        for (int off = 16; off; off >>= 1) sm += __shfl_xor(sm, off, 32);
        const float inv = 1.f / sm;
        if (lane < SRC_LEN)      s_sc[bb][lane]      = e0 * inv;
        if (lane + 32 < SRC_LEN) s_sc[bb][lane + 32] = e1 * inv;
      }
      __syncthreads();

      if (tid < 2 * SRC_LEN) {        // attn probs (last timestep persists)
        const int bi = tid / SRC_LEN, s = tid % SRC_LEN;
        attnF[(blk * 2 + bi) * SRC_LEN + s] = s_sc[bi][s];
      }

      for (int h = sub * 32 + lane; h < HID; h += 64) {
        const float* cb = context + (size_t)b * HID + h;
        float a = 0.f;
#pragma unroll 5
        for (int s = 0; s < SRC_LEN; ++s)
          a += s_sc[bb][s] * cb[(size_t)s * BATCH * HID];
        wctx[b * HID + h] = a;
      }
    }
    grid_barrier(bar);

    // ======= stage 5: out[t] = tanh(cat(wctx, rnn_out) @ Wao^T) =======
    {
      const float* a0 = wctx + m * HID + ko;
      const float* a1 = h1n  + m * HID + ko;
      float* out_t = outputs + (size_t)t * S;

      v8f acc = {};
#pragma unroll 4
      for (int k = 0; k < HID; k += 4)
        acc = wmma4(ld2(a0 + k), ld2(bao + k), acc);
#pragma unroll 4
      for (int k = 0; k < HID; k += 4)
        acc = wmma4(ld2(a1 + k), ld2(bao + HID + k), acc);
#pragma unroll
      for (int r = 0; r < 8; ++r)
        out_t[(wv * 16 + r + 8 * hi) * HID + col] = tanh_fast(acc[r]);
    }
    grid_barrier(bar);  // makes outputs[t] visible to stage 1 of step t+1
  }
}

// ---------------------------------------------------------------------------
// Host driver: init copies + one persistent kernel + final state copies.
// ---------------------------------------------------------------------------
extern "C" void kernel_launch(void* const* d_in, const int* in_sizes, int n_in,
                              void* d_out_v, int out_size, void* d_ws,
                              size_t ws_size, hipStream_t stream)
{
  (void)in_sizes; (void)n_in; (void)out_size; (void)ws_size;

  const int*   tokens   = (const int*)  d_in[0];   // [T, B]
  const float* h0_in    = (const float*)d_in[1];   // [2, B, H]
  const float* c0_in    = (const float*)d_in[2];   // [2, B, H]
  const float* context  = (const float*)d_in[3];   // [SRC, B, H]
  const float* init_out = (const float*)d_in[4];   // [B, H]
  const float* emb      = (const float*)d_in[5];   // [V, E]
  const float* Wih0     = (const float*)d_in[6];
  const float* Whh0     = (const float*)d_in[7];
  const float* b0       = (const float*)d_in[8];
  const float* Wih1     = (const float*)d_in[9];
  const float* Whh1     = (const float*)d_in[10];
  const float* b1       = (const float*)d_in[11];
  const float* Wai      = (const float*)d_in[12];
  const float* Wao      = (const float*)d_in[13];

  float* out     = (float*)d_out_v;
  float* outputs = out;                                   // [T, B, H]
  float* hF      = out + (size_t)T_LEN * BATCH * HID;     // [2, B, H]
  float* cF      = hF + 2 * BATCH * HID;                  // [2, B, H]
  float* attnF   = cF + 2 * BATCH * HID;                  // [B, SRC]

  const int S = BATCH * HID;                              // 32768 floats
  float* ws   = (float*)d_ws;
  float* h0s  = ws;                   // ping-pong pairs: 2*S each
  float* c0s  = h0s + 2 * S;
  float* h1s  = c0s + 2 * S;
  float* c1s  = h1s + 2 * S;
  float* tgt  = c1s + 2 * S;          // S
  float* wctx = tgt + S;              // S
  unsigned* bar = (unsigned*)(wctx + S);  // 2 x u32 barrier state

  const size_t bytesS = (size_t)S * sizeof(float);
  hipMemcpyAsync(h0s, h0_in,     bytesS, hipMemcpyDeviceToDevice, stream);
  hipMemcpyAsync(h1s, h0_in + S, bytesS, hipMemcpyDeviceToDevice, stream);
  hipMemcpyAsync(c0s, c0_in,     bytesS, hipMemcpyDeviceToDevice, stream);
  hipMemcpyAsync(c1s, c0_in + S, bytesS, hipMemcpyDeviceToDevice, stream);
  hipMemsetAsync(bar, 0, 2 * sizeof(unsigned), stream);   // ws is poisoned: must reset

  decoder_kernel<<<dim3(NBLK), dim3(NTHR), 0, stream>>>(
      tokens, emb, init_out, context,
      Wih0, Whh0, b0, Wih1, Whh1, b1, Wai, Wao,
      outputs, attnF,
      h0s, c0s, h1s, c1s, tgt, wctx, bar);

  // Final step (t = 49) wrote state set (49+1)&1 == T_LEN&1 == 0.
  const int fin = T_LEN & 1;
  hipMemcpyAsync(hF,     h0s + fin * S, bytesS, hipMemcpyDeviceToDevice, stream);
  hipMemcpyAsync(hF + S, h1s + fin * S, bytesS, hipMemcpyDeviceToDevice, stream);
  hipMemcpyAsync(cF,     c0s + fin * S, bytesS, hipMemcpyDeviceToDevice, stream);
  hipMemcpyAsync(cF + S, c1s + fin * S, bytesS, hipMemcpyDeviceToDevice, stream);
}